// Matcher_72524817760388
// MI455X (gfx1250) — compile-verified
//
#include <hip/hip_runtime.h>

// Matcher: anchors [N1,4] vs GT boxes [N2,4] (y1,x1,y2,x2 in [0,1]).
// Outputs (concatenated in d_out): matched_vals f32[N1], matched_idx i32[N1],
// labels i32[N1] (1 if iou>=0.7, 0 if iou<0.3, else -1).
//
// VALU-bound: 33.5M IoU pairs (~0.5G f32 ops) vs ~7MB of HBM traffic (~0.3us
// at 23.3 TB/s). Optimizations:
//  - division-free running argmax via cross-multiplication (no v_rcp per pair)
//  - GT boxes + precomputed areas staged in LDS via gfx1250 async copy
//    (global_load_async_to_lds_b128 + s_wait_asynccnt split counter)
//  - 2 anchors per thread: each LDS broadcast read feeds 2 IoU evals, and the
//    two independent argmax select-chains interleave for VOPD dual-issue
//  - constant-trip fast path for full 128-box tiles (kills SALU loop overhead)
//  - no exec divergence in the hot loop (inactive lanes run on a zero box)

#define N2_TILE 128
#define BLOCK   256
#define ITEMS   2   // anchors per thread

__device__ __forceinline__ void iou_step(const float4 a, const float area1,
                                         const float4 b, const float a2,
                                         const int j,
                                         float& bn, float& bd, int& bi) {
    const float iy1 = fmaxf(a.x, b.x);
    const float ix1 = fmaxf(a.y, b.y);
    const float iy2 = fminf(a.z, b.z);
    const float ix2 = fminf(a.w, b.w);
    const float dy  = fmaxf(iy2 - iy1, 0.0f);
    const float dx  = fmaxf(ix2 - ix1, 0.0f);
    const float inter = dy * dx;
    const float uni   = (area1 + a2) - inter;
    // inter/uni > bn/bd  <=>  inter*bd > bn*uni  (both dens >= 0; uni==0
    // implies inter==0 so no spurious update; strict > keeps the FIRST max,
    // matching jnp.argmax tie-breaking).
    const bool better = (inter * bd) > (bn * uni);
    bn = better ? inter : bn;
    bd = better ? uni   : bd;
    bi = better ? j     : bi;
}

__global__ __launch_bounds__(BLOCK) void matcher_iou_kernel(
    const float* __restrict__ boxes1,
    const float* __restrict__ boxes2,
    float* __restrict__ out_vals,
    int*   __restrict__ out_idx,
    int*   __restrict__ out_lab,
    int n1, int n2)
{
    __shared__ float4 sbox[N2_TILE];   // GT boxes (y1,x1,y2,x2)
    __shared__ float  sarea[N2_TILE];  // GT areas

    const int t  = threadIdx.x;
    const int i0 = blockIdx.x * (BLOCK * ITEMS) + t;   // anchor 0
    const int i1 = i0 + BLOCK;                         // anchor 1 (coalesced)

    // Per-anchor boxes; inactive lanes get a zero box (inter==0 => never
    // updates best), so the hot loop runs with full EXEC, no divergence.
    float4 a0 = make_float4(0.f, 0.f, 0.f, 0.f);
    float4 a1 = make_float4(0.f, 0.f, 0.f, 0.f);
    if (i0 < n1) a0 = ((const float4*)boxes1)[i0];
    if (i1 < n1) a1 = ((const float4*)boxes1)[i1];
    const float area0 = (a0.z - a0.x) * (a0.w - a0.y);
    const float area1 = (a1.z - a1.x) * (a1.w - a1.y);

    float bn0 = 0.0f, bd0 = 1.0f;  int bi0 = 0;
    float bn1 = 0.0f, bd1 = 1.0f;  int bi1 = 0;

    const float4* g2 = (const float4*)boxes2;

    for (int jt = 0; jt < n2; jt += N2_TILE) {
        const int cnt = min(n2 - jt, N2_TILE);

        // ---- Stage GT tile into LDS via CDNA5 async global->LDS copy ----
        if (t < cnt) {
            // Low 32 bits of a flat pointer into __shared__ == LDS byte
            // offset (LDS aperture keeps the offset in addr[31:0]).
            unsigned lds_addr = (unsigned)(size_t)(&sbox[t]);
            const float4* gsrc = g2 + (jt + t);
            asm volatile("global_load_async_to_lds_b128 %0, %1, off"
                         :
                         : "v"(lds_addr), "v"(gsrc)
                         : "memory");
        }
        // Split dependency counter for async LDS DMA (no-op for lanes that
        // issued nothing: their ASYNCcnt is already 0).
        asm volatile("s_wait_asynccnt 0" ::: "memory");
        __syncthreads();

        if (t < cnt) {
            const float4 b = sbox[t];
            sarea[t] = (b.z - b.x) * (b.w - b.y);
        }
        __syncthreads();

        // ---- Hot loop: broadcast LDS reads feed 2 anchors per thread ----
        if (cnt == N2_TILE) {
            // Constant trip count: no scalar loop bookkeeping per iteration.
#pragma unroll 8
            for (int j = 0; j < N2_TILE; ++j) {
                const float4 b  = sbox[j];
                const float  a2 = sarea[j];
                iou_step(a0, area0, b, a2, jt + j, bn0, bd0, bi0);
                iou_step(a1, area1, b, a2, jt + j, bn1, bd1, bi1);
            }
        } else {
#pragma unroll 4
            for (int j = 0; j < cnt; ++j) {
                const float4 b  = sbox[j];
                const float  a2 = sarea[j];
                iou_step(a0, area0, b, a2, jt + j, bn0, bd0, bi0);
                iou_step(a1, area1, b, a2, jt + j, bn1, bd1, bi1);
            }
        }
        __syncthreads();
    }

    if (i0 < n1) {
        const float mv = bn0 / fmaxf(bd0, 1e-10f);  // one divide per anchor
        out_vals[i0] = mv;
        out_idx[i0]  = bi0;
        out_lab[i0]  = (mv >= 0.7f) ? 1 : ((mv < 0.3f) ? 0 : -1);
    }
    if (i1 < n1) {
        const float mv = bn1 / fmaxf(bd1, 1e-10f);
        out_vals[i1] = mv;
        out_idx[i1]  = bi1;
        out_lab[i1]  = (mv >= 0.7f) ? 1 : ((mv < 0.3f) ? 0 : -1);
    }
}

extern "C" void kernel_launch(void* const* d_in, const int* in_sizes, int n_in,
                              void* d_out, int out_size, void* d_ws, size_t ws_size,
                              hipStream_t stream) {
    (void)n_in; (void)out_size; (void)d_ws; (void)ws_size;
    const float* boxes1 = (const float*)d_in[0];   // [N1,4] f32
    const float* boxes2 = (const float*)d_in[1];   // [N2,4] f32
    const int n1 = in_sizes[0] / 4;
    const int n2 = in_sizes[1] / 4;

    // d_out = [vals f32 | idx i32 | labels i32], each n1 4-byte elements.
    float* out_vals = (float*)d_out;
    int*   out_idx  = (int*)d_out + n1;
    int*   out_lab  = (int*)d_out + 2 * n1;

    const int per_block = BLOCK * ITEMS;                  // 512 anchors/block
    const int grid      = (n1 + per_block - 1) / per_block;  // 512 blocks
    matcher_iou_kernel<<<grid, BLOCK, 0, stream>>>(boxes1, boxes2,
                                                   out_vals, out_idx, out_lab,
                                                   n1, n2);
}